// EdgeConvBlock_21741124452962
// MI455X (gfx1250) — compile-verified
//
#include <hip/hip_runtime.h>
#include <hip/hip_bf16.h>

typedef __attribute__((ext_vector_type(2))) float v2f;
typedef __attribute__((ext_vector_type(8))) float v8f;

#define NPTS 4096
#define NCH  64
#define NB   8
#define KNN  20
#define NOUT 64
#define NEGINF (-3.402823466e38f)

// ---------------------------------------------------------------------------
// Kernel 0: half squared norms  sqh[b][n] = 0.5 * sum_c x[b][c][n]^2
// ---------------------------------------------------------------------------
__global__ __launch_bounds__(256) void sq_kernel(const float* __restrict__ x,
                                                 float* __restrict__ sqh) {
    int i = blockIdx.x * 256 + threadIdx.x;          // 0 .. B*N-1
    int b = i >> 12, n = i & (NPTS - 1);
    const float* xb = x + (size_t)b * NCH * NPTS + n;
    float s = 0.f;
#pragma unroll
    for (int c = 0; c < NCH; ++c) {
        float v = xb[(size_t)c * NPTS];
        s += v * v;
    }
    sqh[i] = 0.5f * s;
}

// ---------------------------------------------------------------------------
// Kernel 1: fold BN into conv weights.
//   U'[o][c] = inv[o]*(W[o][c]   - W[o][c+64])   (applies to central point)
//   V'[o][c] = inv[o]* W[o][c+64]                (applies to neighbor)
//   bias'[o] = beta[o] - mean[o]*inv[o]
// ---------------------------------------------------------------------------
__global__ __launch_bounds__(256) void fold_kernel(const float* __restrict__ w,
                                                   const float* __restrict__ gamma,
                                                   const float* __restrict__ beta,
                                                   const float* __restrict__ mean,
                                                   const float* __restrict__ var,
                                                   float* __restrict__ Uf,
                                                   float* __restrict__ Vf,
                                                   float* __restrict__ biasf) {
    int i = blockIdx.x * 256 + threadIdx.x;          // 0 .. 4095
    int o = i >> 6, c = i & 63;
    float inv = gamma[o] * rsqrtf(var[o] + 1e-5f);
    float w1 = w[o * 128 + c];
    float w2 = w[o * 128 + 64 + c];
    Uf[o * 64 + c] = inv * (w1 - w2);
    Vf[o * 64 + c] = inv * w2;
    if (c == 0) biasf[o] = beta[o] - mean[o] * inv;
}

// ---------------------------------------------------------------------------
// Kernel 2: P = U' * x, Q = V' * x via WMMA f32 16x16x4, stored transposed
// as [b][n][o] so the gather kernel reads coalesced 64-float columns.
// Block = 256 (8 waves). Each wave does one 16x16 output tile for BOTH P and
// Q (shared B fragments). Grid = B * (N/32).
// ---------------------------------------------------------------------------
__global__ __launch_bounds__(256) void pq_kernel(const float* __restrict__ x,
                                                 const float* __restrict__ Uf,
                                                 const float* __restrict__ Vf,
                                                 float* __restrict__ Pt,
                                                 float* __restrict__ Qt) {
    int b     = blockIdx.x >> 7;                     // N/32 = 128 col groups
    int nBase = (blockIdx.x & 127) * 32;
    int wave  = threadIdx.x >> 5;
    int lane  = threadIdx.x & 31;
    int l16   = lane & 15;
    int half  = lane >> 4;
    int otile = wave & 3;                            // 4 o-tiles of 16
    int colBase = nBase + (wave >> 2) * 16;          // 2 n-tiles per block

    const float* xb = x + (size_t)b * NCH * NPTS;

    // A fragments for U' and V' (A: lane -> M=o, K split over vgpr/lane-half)
    v2f ua[16], va[16];
#pragma unroll
    for (int kk = 0; kk < 16; ++kk) {
        int c0 = kk * 4 + half * 2;
        int o  = otile * 16 + l16;
        ua[kk].x = Uf[o * 64 + c0];
        ua[kk].y = Uf[o * 64 + c0 + 1];
        va[kk].x = Vf[o * 64 + c0];
        va[kk].y = Vf[o * 64 + c0 + 1];
    }

    v8f cp = {};
    v8f cq = {};
#pragma unroll
    for (int kk = 0; kk < 16; ++kk) {
        int c0 = kk * 4 + half * 2;
        v2f bf;
        bf.x = xb[(size_t)c0 * NPTS + colBase + l16];
        bf.y = xb[(size_t)(c0 + 1) * NPTS + colBase + l16];
        cp = __builtin_amdgcn_wmma_f32_16x16x4_f32(false, ua[kk], false, bf,
                                                   (short)0, cp, false, false);
        cq = __builtin_amdgcn_wmma_f32_16x16x4_f32(false, va[kk], false, bf,
                                                   (short)0, cq, false, false);
    }

    // C layout: vgpr v -> row M = v + half*8, lane -> col N = l16
#pragma unroll
    for (int v = 0; v < 8; ++v) {
        int o = otile * 16 + v + half * 8;
        int n = colBase + l16;
        size_t ad = ((size_t)b * NPTS + n) * NOUT + o;
        Pt[ad] = cp[v];
        Qt[ad] = cq[v];
    }
}

// ---------------------------------------------------------------------------
// Kernel 3: fused Gram-matrix + top-K neighbors.
// Block = 512 threads (16 waves), handles 16 rows of one batch.
// Each wave owns a 256-column stripe: 16 WMMA score tiles, accumulator
// pre-loaded with -0.5*||x_j||^2 so scores rank identically to -dist.
// Scores pass through a per-wave LDS tile so lanes 0-15 each own one row and
// keep a private top-20; the 16 per-wave partial lists are merged at the end.
// ---------------------------------------------------------------------------
__global__ __launch_bounds__(512) void knn_kernel(const float* __restrict__ x,
                                                  const float* __restrict__ sqh,
                                                  int* __restrict__ knn_out) {
    __shared__ float tileS[16][16 * 17];             // per-wave 16x16 (+pad)
    __shared__ float mergeV[256][KNN];               // 16 waves * 16 rows
    __shared__ int   mergeI[256][KNN];

    int b       = blockIdx.x >> 8;                   // N/16 = 256 row tiles
    int rowBase = (blockIdx.x & 255) << 4;
    int wave    = threadIdx.x >> 5;
    int lane    = threadIdx.x & 31;
    int l16     = lane & 15;
    int half    = lane >> 4;

    const float* xb = x + (size_t)b * NCH * NPTS;

    // A fragments: the 16 rows' features (reused across all 16 column tiles)
    v2f afrag[16];
#pragma unroll
    for (int kk = 0; kk < 16; ++kk) {
        int c0 = kk * 4 + half * 2;
        afrag[kk].x = xb[(size_t)c0 * NPTS + rowBase + l16];
        afrag[kk].y = xb[(size_t)(c0 + 1) * NPTS + rowBase + l16];
    }

    // per-lane private top-K (lanes 0-15 own rows rowBase+lane)
    float kv[KNN];
    int   ki[KNN];
#pragma unroll
    for (int t = 0; t < KNN; ++t) { kv[t] = NEGINF; ki[t] = 0; }
    float vmin = NEGINF;
    int   pmin = 0;

    int colStart = wave * 256;
    float* tw = &tileS[wave][0];

    for (int t = 0; t < 16; ++t) {
        int colBase = colStart + t * 16;

        // score = x_i . x_j - 0.5*||x_j||^2  (per-row-constant terms dropped)
        float cinit = -sqh[b * NPTS + colBase + l16];
        v8f c;
#pragma unroll
        for (int v = 0; v < 8; ++v) c[v] = cinit;

#pragma unroll
        for (int kk = 0; kk < 16; ++kk) {
            int c0 = kk * 4 + half * 2;
            v2f bf;
            bf.x = xb[(size_t)c0 * NPTS + colBase + l16];
            bf.y = xb[(size_t)(c0 + 1) * NPTS + colBase + l16];
            c = __builtin_amdgcn_wmma_f32_16x16x4_f32(false, afrag[kk], false,
                                                      bf, (short)0, c, false,
                                                      false);
        }

        // stage tile to LDS (row-major with pad 17 to dodge bank conflicts)
#pragma unroll
        for (int v = 0; v < 8; ++v)
            tw[(v + half * 8) * 17 + l16] = c[v];
        __syncthreads();

        // transposed consume: lane r scans its row's 16 new candidates
        if (lane < 16) {
#pragma unroll
            for (int j = 0; j < 16; ++j) {
                float val = tw[lane * 17 + j];
                if (val > vmin) {
                    kv[pmin] = val;
                    ki[pmin] = colBase + j;
                    vmin = kv[0]; pmin = 0;
#pragma unroll
                    for (int q = 1; q < KNN; ++q)
                        if (kv[q] < vmin) { vmin = kv[q]; pmin = q; }
                }
            }
        }
        __syncthreads();
    }

    // publish per-wave partial top-K lists
    if (lane < 16) {
        int slot = wave * 16 + lane;
#pragma unroll
        for (int t = 0; t < KNN; ++t) {
            mergeV[slot][t] = kv[t];
            mergeI[slot][t] = ki[t];
        }
    }
    __syncthreads();

    // final merge: wave r reduces row r's 320 candidates to global top-20
    {
        int r = wave;                                 // 16 waves, 16 rows
        float cv[10];
        int   ci[10];
#pragma unroll
        for (int u = 0; u < 10; ++u) {
            int m  = lane * 10 + u;                   // 320 candidates
            int w2 = m / KNN;
            int tt = m % KNN;
            cv[u] = mergeV[w2 * 16 + r][tt];
            ci[u] = mergeI[w2 * 16 + r][tt];
        }
        int* outp = knn_out + ((size_t)b * NPTS + rowBase + r) * KNN;
        for (int it = 0; it < KNN; ++it) {
            // local arg-max over this lane's 10 slots
            float bv = cv[0];
            int   bu = 0;
#pragma unroll
            for (int u = 1; u < 10; ++u)
                if (cv[u] > bv) { bv = cv[u]; bu = u; }
            // wave arg-max via xor shuffles
            float rv = bv;
            int   rl = lane;
#pragma unroll
            for (int off = 16; off > 0; off >>= 1) {
                float ov = __shfl_xor(rv, off, 32);
                int   ol = __shfl_xor(rl, off, 32);
                if (ov > rv || (ov == rv && ol < rl)) { rv = ov; rl = ol; }
            }
            int myIdx = ci[bu];
            int sel   = __shfl(myIdx, rl, 32);
            if (lane == 0) outp[it] = sel;
            if (lane == rl) cv[bu] = NEGINF;
        }
    }
}

// ---------------------------------------------------------------------------
// Kernel 4: gather + max over K + bias + ReLU.
//   out[b][o][n] = relu(P[b][n][o] + max_k Q[b][nbr_k][o] + bias[o])
// (max/relu commute; P and bias are k-invariant.)  One wave per point,
// lanes cover o = lane and lane+32; Q reads are coalesced 128B halves.
// ---------------------------------------------------------------------------
__global__ __launch_bounds__(256) void gather_kernel(const float* __restrict__ Pt,
                                                     const float* __restrict__ Qt,
                                                     const int* __restrict__ knn_in,
                                                     const float* __restrict__ biasf,
                                                     float* __restrict__ out) {
    int wave = threadIdx.x >> 5;
    int lane = threadIdx.x & 31;
    int gp   = blockIdx.x * 8 + wave;                // 0 .. B*N-1
    int b    = gp >> 12;
    int n    = gp & (NPTS - 1);

    const int* kn = knn_in + (size_t)gp * KNN;
    int o0 = lane, o1 = lane + 32;
    float m0 = NEGINF, m1 = NEGINF;
#pragma unroll
    for (int k = 0; k < KNN; ++k) {
        int idx = kn[k];
        const float* q = Qt + ((size_t)b * NPTS + idx) * NOUT;
        m0 = fmaxf(m0, q[o0]);
        m1 = fmaxf(m1, q[o1]);
    }
    const float* p = Pt + (size_t)gp * NOUT;
    float r0 = fmaxf(p[o0] + m0 + biasf[o0], 0.f);
    float r1 = fmaxf(p[o1] + m1 + biasf[o1], 0.f);
    out[((size_t)b * NOUT + o0) * NPTS + n] = r0;
    out[((size_t)b * NOUT + o1) * NPTS + n] = r1;
}

// ---------------------------------------------------------------------------
extern "C" void kernel_launch(void* const* d_in, const int* in_sizes, int n_in,
                              void* d_out, int out_size, void* d_ws, size_t ws_size,
                              hipStream_t stream) {
    const float* x     = (const float*)d_in[0];   // (B, C, N)
    const float* convw = (const float*)d_in[1];   // (O, 2C)
    const float* gamma = (const float*)d_in[2];
    const float* beta  = (const float*)d_in[3];
    const float* mean  = (const float*)d_in[4];
    const float* var   = (const float*)d_in[5];
    float* out = (float*)d_out;                   // (B, O, N)

    // workspace layout (bytes)
    char* ws = (char*)d_ws;
    float* sqh   = (float*)(ws);                          // 32768 f  = 128 KB
    int*   knnW  = (int*)  (ws + 131072);                 // 655360 i = 2.5 MB
    float* Uf    = (float*)(ws + 2752512);                // 4096 f
    float* Vf    = (float*)(ws + 2768896);                // 4096 f
    float* biasf = (float*)(ws + 2785280);                // 64 f
    float* Pt    = (float*)(ws + 2785536);                // 2M f = 8 MB
    float* Qt    = (float*)(ws + 11174144);               // 2M f = 8 MB

    sq_kernel    <<<(NB * NPTS) / 256, 256, 0, stream>>>(x, sqh);
    fold_kernel  <<<(NOUT * NCH) / 256, 256, 0, stream>>>(convw, gamma, beta,
                                                          mean, var, Uf, Vf, biasf);
    pq_kernel    <<<NB * (NPTS / 32), 256, 0, stream>>>(x, Uf, Vf, Pt, Qt);
    knn_kernel   <<<NB * (NPTS / 16), 512, 0, stream>>>(x, sqh, knnW);
    gather_kernel<<<(NB * NPTS) / 8, 256, 0, stream>>>(Pt, Qt, knnW, biasf, out);
}